// MonotonicAlignmentLoss_18502719111780
// MI455X (gfx1250) — compile-verified
//
#include <hip/hip_runtime.h>
#include <hip/hip_bf16.h>
#include <stdint.h>

#define B_DIM 64
#define MEL   2048
#define TXT   512
#define ROWS  (B_DIM * MEL)            // 131072 rows, one argmax each
#define WAVES_PER_BLOCK 8
#define ROWS_PER_WAVE   8
#define ROWS_PER_BLOCK  (WAVES_PER_BLOCK * ROWS_PER_WAVE)   // 64
#define NBLOCKS         (ROWS / ROWS_PER_BLOCK)             // 2048

// Issue 4 async b128 copies: one 2KB row (512 f32) global -> LDS.
// Per CDNA5 ISA the offset is added to the LDS and the global address alike,
// so the instruction offset advances BOTH addresses; lanes cover 16B slices.
// th:TH_LOAD_NT: the 256MB tensor is streamed once (> 192MB L2) -> don't cache it,
// keeping L2 free for the reused peaks array between passes.
__device__ __forceinline__ void async_row_prefetch(const float* row_lane_ptr,
                                                   uint32_t lds_lane_byte) {
    asm volatile(
        "global_load_async_to_lds_b128 %0, %1, off th:TH_LOAD_NT\n\t"
        "global_load_async_to_lds_b128 %0, %1, off offset:512 th:TH_LOAD_NT\n\t"
        "global_load_async_to_lds_b128 %0, %1, off offset:1024 th:TH_LOAD_NT\n\t"
        "global_load_async_to_lds_b128 %0, %1, off offset:1536 th:TH_LOAD_NT"
        :: "v"(lds_lane_byte), "v"(row_lane_ptr) : "memory");
}

__global__ void __launch_bounds__(256)
peaks_kernel(const float* __restrict__ attn, int* __restrict__ pk) {
    __shared__ __align__(16) float smem[WAVES_PER_BLOCK * 2 * TXT];  // 32 KB
    const int lane = threadIdx.x & 31;
    const int warp = threadIdx.x >> 5;

    float* buf0 = &smem[warp * (2 * TXT)];
    float* buf1 = buf0 + TXT;
    // Low 32 bits of the flat LDS address == LDS byte offset (aperture layout).
    const uint32_t lds0 = (uint32_t)(uintptr_t)buf0 + (uint32_t)lane * 16u;
    const uint32_t lds1 = lds0 + TXT * 4u;

    const long waveId  = (long)blockIdx.x * WAVES_PER_BLOCK + warp;
    const long rowBase = waveId * ROWS_PER_WAVE;

    // Prime the pipeline: row 0 -> buf0
    async_row_prefetch(attn + rowBase * TXT + lane * 4, lds0);

    for (int r = 0; r < ROWS_PER_WAVE; ++r) {
        if (r + 1 < ROWS_PER_WAVE) {
            // Prefetch next row into the other buffer, then wait until the
            // current row's 4 loads (issued earlier, in-order) are complete.
            async_row_prefetch(attn + (rowBase + r + 1) * TXT + lane * 4,
                               (r & 1) ? lds0 : lds1);
            asm volatile("s_wait_asynccnt 4" ::: "memory");
        } else {
            asm volatile("s_wait_asynccnt 0" ::: "memory");
        }

        const float* buf = (r & 1) ? buf1 : buf0;
        float bestv = -__builtin_inff();
        int   besti = 0;
#pragma unroll
        for (int c = 0; c < 4; ++c) {
            const float4 v = *reinterpret_cast<const float4*>(buf + c * 128 + lane * 4);
            const int base = c * 128 + lane * 4;
            // strict '>' keeps the first (lowest-index) max within the lane
            if (v.x > bestv) { bestv = v.x; besti = base;     }
            if (v.y > bestv) { bestv = v.y; besti = base + 1; }
            if (v.z > bestv) { bestv = v.z; besti = base + 2; }
            if (v.w > bestv) { bestv = v.w; besti = base + 3; }
        }
        // wave32 cross-lane argmax, tie-break on lower index (matches jnp.argmax)
#pragma unroll
        for (int off = 16; off >= 1; off >>= 1) {
            float ov = __shfl_xor(bestv, off, 32);
            int   oi = __shfl_xor(besti, off, 32);
            if (ov > bestv || (ov == bestv && oi < besti)) { bestv = ov; besti = oi; }
        }
        if (lane == 0) pk[rowBase + r] = besti;
    }
}

__global__ void __launch_bounds__(256)
viol_kernel(const int* __restrict__ pk, int* __restrict__ acc) {
    const int total = B_DIM * (MEL - 1);
    int sum = 0;
    for (int t = blockIdx.x * blockDim.x + threadIdx.x; t < total;
         t += gridDim.x * blockDim.x) {
        const int b = t / (MEL - 1);
        const int i = t - b * (MEL - 1);
        const int d = pk[b * MEL + i] - pk[b * MEL + i + 1];
        sum += (d > 0) ? d : 0;
    }
#pragma unroll
    for (int off = 16; off >= 1; off >>= 1) sum += __shfl_xor(sum, off, 32);
    if ((threadIdx.x & 31) == 0 && sum != 0) atomicAdd(acc, sum);  // exact int sum -> deterministic
}

__global__ void zero_kernel(int* acc) { *acc = 0; }

__global__ void finalize_kernel(const int* __restrict__ acc, float* __restrict__ out) {
    out[0] = (float)((double)(*acc) * (1.0 / (double)(B_DIM * MEL)));
}

extern "C" void kernel_launch(void* const* d_in, const int* in_sizes, int n_in,
                              void* d_out, int out_size, void* d_ws, size_t ws_size,
                              hipStream_t stream) {
    (void)in_sizes; (void)n_in; (void)out_size; (void)ws_size;
    const float* attn = (const float*)d_in[0];
    float* out = (float*)d_out;
    int* pk  = (int*)d_ws;        // 131072 ints = 512 KB
    int* acc = pk + ROWS;         // 1 int accumulator

    zero_kernel<<<1, 1, 0, stream>>>(acc);
    peaks_kernel<<<NBLOCKS, 256, 0, stream>>>(attn, pk);
    viol_kernel<<<256, 256, 0, stream>>>(pk, acc);
    finalize_kernel<<<1, 1, 0, stream>>>(acc, out);
}